// GAE_66236985638999
// MI455X (gfx1250) — compile-verified
//
#include <hip/hip_runtime.h>
#include <stdint.h>
#include <math.h>

// ---------------- GAE on gfx1250 (MI455X), wave32 ----------------
// Affine-map scan:  m_t = (c_t, delta_t),  (a1,b1)∘(a2,b2) = (a1*a2, b1 + a1*b2)
//   c_t     = GAMMA*LAMDA*(1-done_t)
//   delta_t = r_t - v_t + GAMMA*(1-done_t)*v_{t+1}
// adv_t = apply(m_t ∘ m_{t+1} ∘ ... ∘ m_{T-1}, 0)

#define GAMMA 0.99f
#define LAMDA 0.95f
#define EPSN  1e-8

#define WAVE   32
#define WPB    8            // waves per block (256 threads)
#define ITERS  64           // 32-wide tiles per wave segment
#define SEG    (WAVE*ITERS) // 2048 elements per wave segment
#define MAX_SEG_LDS 4096    // pass-2 LDS staging capacity (32KB)
#define PFD    4            // prefetch distance (tiles) for backward stream

#define AS1 __attribute__((address_space(1)))
#define AS3 __attribute__((address_space(3)))

// ---- CDNA5 Tensor Data Mover staging (guarded; this toolchain = 6-arg form) ----
#if defined(__has_builtin)
#if __has_builtin(__builtin_amdgcn_tensor_load_to_lds)
#define HAVE_TDM 1
#endif
#if __has_builtin(__builtin_amdgcn_s_wait_tensorcnt)
#define HAVE_TENSORCNT 1
#endif
#endif

#ifdef HAVE_TDM
typedef unsigned int u32x4 __attribute__((ext_vector_type(4)));
typedef int          i32x8 __attribute__((ext_vector_type(8)));
typedef int          i32x4 __attribute__((ext_vector_type(4)));

__device__ inline unsigned int lds_byte_offset(const void* p) {
  return (unsigned int)(unsigned long long)(AS3 const void*)p;
}

// 1-row f32 tensor of n elements (n <= 65535), global -> LDS via TDM.
__device__ inline void tdm_load_row_f32(const float* gsrc, const void* lds_dst,
                                        unsigned int n) {
  unsigned long long ga = (unsigned long long)(const void*)gsrc;
  u32x4 g0;
  g0[0] = 1u;                                        // count=1 (user descriptor)
  g0[1] = lds_byte_offset(lds_dst);                  // lds_addr (bytes)
  g0[2] = (unsigned int)(ga & 0xFFFFFFFFu);          // global_addr[31:0]
  g0[3] = (unsigned int)((ga >> 32) & 0x01FFFFFFull) // global_addr[56:32]
        | (2u << 30);                                // type = 2 ("image")
  i32x8 g1;
  g1[0] = (int)(2u << 16);                           // data_size = 4B
  g1[1] = (int)((n & 0xFFFFu) << 16);                // tensor_dim0[15:0]
  g1[2] = (int)(((n >> 16) & 0xFFFFu) | (1u << 16)); // tensor_dim0[31:16], tensor_dim1=1
  g1[3] = (int)((n & 0xFFFFu) << 16);                // tile_dim0 = n
  g1[4] = 0;                                         // tile_dim1/2 unused
  g1[5] = (int)n;                                    // tensor_dim0_stride = n
  g1[6] = 0;
  g1[7] = 0;
  i32x4 z4; z4[0]=0; z4[1]=0; z4[2]=0; z4[3]=0;      // groups 2/3 unused (<=2D)
  i32x8 z8; z8[0]=0; z8[1]=0; z8[2]=0; z8[3]=0;
  z8[4]=0; z8[5]=0; z8[6]=0; z8[7]=0;
  // 6-arg form on this toolchain (clang-23 / therock-10.0 headers)
  __builtin_amdgcn_tensor_load_to_lds(g0, g1, z4, z4, z8, 0);
}
#endif // HAVE_TDM

// ---------------- Pass 1: per-segment affine aggregates ----------------
__global__ __launch_bounds__(256) void gae_p1_aggregate(
    const float* __restrict__ rew, const float* __restrict__ vp,
    const int* __restrict__ done, long long T, int nseg,
    float* __restrict__ segA, float* __restrict__ segB)
{
  const int lane = threadIdx.x & (WAVE - 1);
  const int seg  = blockIdx.x * WPB + (threadIdx.x >> 5);
  if (seg >= nseg) return;
  const long long base = (long long)seg * SEG;

  long long after = base + SEG;
  float v_hi = (after < T) ? vp[after] : 0.f;  // v_pred just past current tile
  float Ar = 1.f, Br = 0.f;                    // running suffix composition

  for (int it = ITERS - 1; it >= 0; --it) {
    if (lane == 0 && it >= PFD) {              // backward-stream prefetch
      const long long p = base + (long long)(it - PFD) * WAVE;
      __builtin_prefetch(rew  + p, 0, 1);
      __builtin_prefetch(vp   + p, 0, 1);
      __builtin_prefetch(done + p, 0, 1);
    }
    const long long idx = base + (long long)it * WAVE + lane;
    const bool valid = idx < T;
    float r = 0.f, v = 0.f; int d = 1;
    if (valid) { r = rew[idx]; v = vp[idx]; d = done[idx]; }
    float vn = __shfl_down(v, 1, WAVE);
    if (lane == WAVE - 1) vn = v_hi;
    float a = 1.f, b = 0.f;
    if (valid) {
      const float nd = 1.f - (float)d;
      a = (GAMMA * LAMDA) * nd;
      b = r - v + GAMMA * nd * vn;
    }
    v_hi = __shfl(v, 0, WAVE);                 // next (lower) tile's boundary v

    // Kogge-Stone inclusive *suffix* scan of maps across the wave
    #pragma unroll
    for (int dl = 1; dl < WAVE; dl <<= 1) {
      float a2 = __shfl_down(a, dl, WAVE);
      float b2 = __shfl_down(b, dl, WAVE);
      if (lane + dl < WAVE) { b = fmaf(a, b2, b); a *= a2; }
    }
    // lane 0 holds the tile composition; fold into running map (tile ∘ running)
    const float a0 = __shfl(a, 0, WAVE);
    const float b0 = __shfl(b, 0, WAVE);
    Br = fmaf(a0, Br, b0);
    Ar *= a0;
  }
  if (lane == 0) { segA[seg] = Ar; segB[seg] = Br; }
}

// ---------------- Pass 2: scan segment maps -> per-segment incoming adv ----------------
__global__ __launch_bounds__(256) void gae_p2_segscan(
    const float* __restrict__ segA, const float* __restrict__ segB,
    int nseg, float* __restrict__ segIn)
{
  __shared__ float lA[MAX_SEG_LDS];
  __shared__ float lB[MAX_SEG_LDS];
  __shared__ float sa[256], sb[256];
  const int tid = threadIdx.x;
  const bool useLds = (nseg <= MAX_SEG_LDS);

  if (useLds) {
#ifdef HAVE_TDM
    if (tid < WAVE) {                          // one wave issues the two DMAs
      tdm_load_row_f32(segA, lA, (unsigned int)nseg);
      tdm_load_row_f32(segB, lB, (unsigned int)nseg);
#ifdef HAVE_TENSORCNT
      __builtin_amdgcn_s_wait_tensorcnt(0);
#endif
    }
#else
    for (int i = tid; i < nseg; i += 256) { lA[i] = segA[i]; lB[i] = segB[i]; }
#endif
    __syncthreads();
  }

  const int Q  = (nseg + 255) / 256;
  const int s0 = tid * Q;

  // Phase 1: compose this thread's range left -> right
  float A = 1.f, B = 0.f;
  for (int k = 0; k < Q; ++k) {
    const int i = s0 + k;
    if (i >= nseg) break;
    const float a = useLds ? lA[i] : segA[i];
    const float b = useLds ? lB[i] : segB[i];
    B = fmaf(A, b, B);                          // (A,B) ∘ (a,b)
    A *= a;
  }

  // Phase 2: block-wide inclusive suffix scan of thread maps (LDS Kogge-Stone)
  sa[tid] = A; sb[tid] = B;
  __syncthreads();
  for (int d = 1; d < 256; d <<= 1) {
    float a2 = 1.f, b2 = 0.f;
    if (tid + d < 256) { a2 = sa[tid + d]; b2 = sb[tid + d]; }
    __syncthreads();
    B = fmaf(A, b2, B);
    A *= a2;
    sa[tid] = A; sb[tid] = B;
    __syncthreads();
  }
  // Exclusive suffix = inclusive of (tid+1); identity at the end
  float cA = 1.f, cB = 0.f;
  if (tid + 1 < 256) { cA = sa[tid + 1]; cB = sb[tid + 1]; }

  // Phase 3: replay right -> left; boundary value after seg i = b-part of suffix
  for (int k = Q - 1; k >= 0; --k) {
    const int i = s0 + k;
    if (i >= nseg) continue;
    segIn[i] = cB;                              // apply(carry, adv_after_T=0) = cB
    const float a = useLds ? lA[i] : segA[i];
    const float b = useLds ? lB[i] : segB[i];
    cB = fmaf(a, cB, b);                        // carry = m_i ∘ carry
    cA *= a;
  }
}

// ---------------- Pass 3: replay, write adv, deterministic partial sums ----------------
__global__ __launch_bounds__(256) void gae_p3_replay(
    const float* __restrict__ rew, const float* __restrict__ vp,
    const int* __restrict__ done, const float* __restrict__ segIn,
    long long T, int nseg, float* __restrict__ out,
    double* __restrict__ blkSums)
{
  __shared__ double sS[WPB], sQ[WPB];
  const int lane = threadIdx.x & (WAVE - 1);
  const int wv   = threadIdx.x >> 5;
  const int seg  = blockIdx.x * WPB + wv;

  double lsum = 0.0, lsq = 0.0;
  if (seg < nseg) {
    const long long base = (long long)seg * SEG;
    long long after = base + SEG;
    float v_hi = (after < T) ? vp[after] : 0.f;
    float carry = segIn[seg];                   // adv at segment's right boundary

    for (int it = ITERS - 1; it >= 0; --it) {
      if (lane == 0 && it >= PFD) {
        const long long p = base + (long long)(it - PFD) * WAVE;
        __builtin_prefetch(rew  + p, 0, 1);
        __builtin_prefetch(vp   + p, 0, 1);
        __builtin_prefetch(done + p, 0, 1);
      }
      const long long idx = base + (long long)it * WAVE + lane;
      const bool valid = idx < T;
      float r = 0.f, v = 0.f; int d = 1;
      if (valid) { r = rew[idx]; v = vp[idx]; d = done[idx]; }
      float vn = __shfl_down(v, 1, WAVE);
      if (lane == WAVE - 1) vn = v_hi;
      float a = 1.f, b = 0.f;
      if (valid) {
        const float nd = 1.f - (float)d;
        a = (GAMMA * LAMDA) * nd;
        b = r - v + GAMMA * nd * vn;
      }
      v_hi = __shfl(v, 0, WAVE);

      #pragma unroll
      for (int dl = 1; dl < WAVE; dl <<= 1) {
        float a2 = __shfl_down(a, dl, WAVE);
        float b2 = __shfl_down(b, dl, WAVE);
        if (lane + dl < WAVE) { b = fmaf(a, b2, b); a *= a2; }
      }
      const float adv = fmaf(a, carry, b);      // suffix map applied to boundary
      if (valid) {
        out[idx] = adv;
        lsum += (double)adv;
        lsq  += (double)adv * (double)adv;
      }
      carry = __shfl(adv, 0, WAVE);             // adv at tile start = next boundary
    }
  }

  // wave reduce (fixed order), then block reduce (fixed order) -> deterministic
  #pragma unroll
  for (int dl = 16; dl > 0; dl >>= 1) {
    lsum += __shfl_down(lsum, dl, WAVE);
    lsq  += __shfl_down(lsq,  dl, WAVE);
  }
  if (lane == 0) { sS[wv] = lsum; sQ[wv] = lsq; }
  __syncthreads();
  if (threadIdx.x == 0) {
    double s = 0.0, q = 0.0;
    #pragma unroll
    for (int w = 0; w < WPB; ++w) { s += sS[w]; q += sQ[w]; }
    blkSums[2 * blockIdx.x]     = s;
    blkSums[2 * blockIdx.x + 1] = q;
  }
}

// ---------------- Pass 4: mean / inv(std+eps) ----------------
__global__ __launch_bounds__(256) void gae_p4_stats(
    const double* __restrict__ blkSums, int nblk, long long T,
    float* __restrict__ stats)
{
  __shared__ double sS[256], sQ[256];
  double s = 0.0, q = 0.0;
  for (int i = threadIdx.x; i < nblk; i += 256) {
    s += blkSums[2 * i];
    q += blkSums[2 * i + 1];
  }
  sS[threadIdx.x] = s; sQ[threadIdx.x] = q;
  __syncthreads();
  for (int d = 128; d > 0; d >>= 1) {
    if (threadIdx.x < d) {
      sS[threadIdx.x] += sS[threadIdx.x + d];
      sQ[threadIdx.x] += sQ[threadIdx.x + d];
    }
    __syncthreads();
  }
  if (threadIdx.x == 0) {
    const double mean = sS[0] / (double)T;
    double var = sQ[0] / (double)T - mean * mean;
    if (var < 0.0) var = 0.0;
    stats[0] = (float)mean;
    stats[1] = (float)(1.0 / (sqrt(var) + EPSN));
  }
}

// ---------------- Pass 5: in-place normalize (float4) ----------------
__global__ __launch_bounds__(256) void gae_p5_norm(
    float* __restrict__ out, long long T, long long n4,
    const float* __restrict__ stats)
{
  const float mean = stats[0];
  const float inv  = stats[1];
  float4* o4 = (float4*)out;
  const long long stride = (long long)gridDim.x * blockDim.x;
  for (long long i = (long long)blockIdx.x * blockDim.x + threadIdx.x;
       i < n4; i += stride) {
    float4 v = o4[i];
    v.x = (v.x - mean) * inv;
    v.y = (v.y - mean) * inv;
    v.z = (v.z - mean) * inv;
    v.w = (v.w - mean) * inv;
    o4[i] = v;
  }
  if (blockIdx.x == 0 && threadIdx.x == 0) {
    for (long long j = n4 * 4; j < T; ++j) out[j] = (out[j] - mean) * inv;
  }
}

// ---------------- host-side launcher ----------------
extern "C" void kernel_launch(void* const* d_in, const int* in_sizes, int n_in,
                              void* d_out, int out_size, void* d_ws, size_t ws_size,
                              hipStream_t stream) {
  const float* rewards = (const float*)d_in[0];
  const float* vpred   = (const float*)d_in[1];
  const int*   done    = (const int*)d_in[2];
  float* out = (float*)d_out;

  const long long T = (long long)in_sizes[0];
  const int nseg = (int)((T + SEG - 1) / SEG);          // 4096 for T=8.4M
  const int nblk = (nseg + WPB - 1) / WPB;              // 512

  // workspace layout: segA|segB|segIn (f32), blkSums (f64 pairs), stats (f32 x2)
  char* ws = (char*)d_ws;
  float* segA  = (float*)ws;
  float* segB  = segA + nseg;
  float* segIn = segB + nseg;
  size_t off = (((size_t)3 * (size_t)nseg * sizeof(float)) + 15) & ~(size_t)15;
  double* blkSums = (double*)(ws + off);
  float*  stats   = (float*)(ws + off + (size_t)nblk * 2 * sizeof(double));

  gae_p1_aggregate<<<nblk, 256, 0, stream>>>(rewards, vpred, done, T, nseg,
                                             segA, segB);
  gae_p2_segscan<<<1, 256, 0, stream>>>(segA, segB, nseg, segIn);
  gae_p3_replay<<<nblk, 256, 0, stream>>>(rewards, vpred, done, segIn, T, nseg,
                                          out, blkSums);
  gae_p4_stats<<<1, 256, 0, stream>>>(blkSums, nblk, T, stats);

  const long long n4 = T / 4;
  int gnorm = (int)((n4 + 255) / 256);
  if (gnorm < 1) gnorm = 1;
  gae_p5_norm<<<gnorm, 256, 0, stream>>>(out, T, n4, stats);
}